// DynamicFilterModel_8796093022247
// MI455X (gfx1250) — compile-verified
//
#include <hip/hip_runtime.h>
#include <hip/hip_bf16.h>
#include <math.h>

typedef __attribute__((ext_vector_type(2))) float v2f;
typedef __attribute__((ext_vector_type(8))) float v8f;
typedef __attribute__((ext_vector_type(4))) unsigned int v4u;
typedef __attribute__((ext_vector_type(8))) int v8i;
typedef __attribute__((ext_vector_type(4))) int v4i;

#define Bn   16
#define Hh   224
#define Wn   224
#define CINc 8
#define Ff   256
#define NCc  16
#define NPp  1152   // CIN*NC*3*3
#define HW   (Hh * Wn)

__device__ __forceinline__ float elu1(float x) {
    return x > 0.0f ? x : expm1f(x);
}

// A-fragment LDS float-offset (constant part) for WMMA step kk.
// K ordering is (kh, kw, cin): k' = kh*24 + kw*8 + cin.
// Window layout: win[kh][col][cin], cin fastest -> offset = (kh*18+kw)*8 + cin
// (plus per-lane n*8 + 2*tg added via the base pointer).
__device__ __forceinline__ constexpr int aoff(int kk) {
    const int idx = 4 * kk;            // first k' of this step (cin part 0 or 4)
    const int kh = idx / 24;
    const int rem = idx % 24;
    const int kw = rem / 8;
    const int cinb = rem % 8;          // 0 or 4
    return (kh * 18 + kw) * CINc + cinb;
}

// h = elu(filter_inpt @ W1 + b1)   [16,256]
__global__ void dfn_param_kernel(const float* __restrict__ fin,
                                 const float* __restrict__ W1,
                                 const float* __restrict__ b1,
                                 float* __restrict__ h) {
    int id = blockIdx.x * blockDim.x + threadIdx.x;  // B*F = 4096 threads
    int b = id >> 8;
    int f = id & 255;
    float s = b1[f];
    const float* hr = fin + b * Ff;
#pragma unroll 8
    for (int k = 0; k < Ff; ++k) s = fmaf(hr[k], W1[k * Ff + f], s);
    h[id] = elu1(s);
}

// wdyn[b] in WMMA-ready layout [kpair(36)][c(16)][klo(2)], K permuted to
// (kh,kw,cin) order;  biasbuf = h @ Wb + bb + dfn_bias  [16,16]
__global__ void dfn_dyn_kernel(const float* __restrict__ h,
                               const float* __restrict__ Wd,
                               const float* __restrict__ bd,
                               const float* __restrict__ Wb,
                               const float* __restrict__ bb,
                               const float* __restrict__ dfnb,
                               float* __restrict__ wdyn,
                               float* __restrict__ biasbuf) {
    int id = blockIdx.x * blockDim.x + threadIdx.x;
    if (id < Bn * NPp) {
        const int b = id / NPp;
        const int x = id - b * NPp;
        const int kpair = x >> 5;          // 0..35
        const int c     = (x >> 1) & 15;   // out channel
        const int klo   = x & 1;
        const int kprime = kpair * 2 + klo;        // 0..71, (kh,kw,cin) order
        const int kh  = kprime / 24;
        const int rem = kprime % 24;
        const int kw  = rem / 8;
        const int cin = rem % 8;
        // original dyn column: p = cin*9 + kh*3 + kw, j = p*16 + c
        const int j = (cin * 9 + kh * 3 + kw) * NCc + c;
        float s = bd[j];
        const float* hr = h + b * Ff;
#pragma unroll 8
        for (int k = 0; k < Ff; ++k) s = fmaf(hr[k], Wd[k * NPp + j], s);
        wdyn[id] = s;
    } else {
        const int id2 = id - Bn * NPp;
        if (id2 < Bn * NCc) {
            const int b = id2 >> 4;
            const int c = id2 & 15;
            float s = bb[c];
            const float* hr = h + b * Ff;
#pragma unroll 8
            for (int k = 0; k < Ff; ++k) s = fmaf(hr[k], Wb[k * NCc + c], s);
            biasbuf[id2] = s + dfnb[c];
        }
    }
}

// Per-sample dynamic 3x3x8->16 conv as [HW,72]@[72,16] tiles via f32 WMMA.
// Weights come in via the Tensor Data Mover (1-D D#), halo windows via
// global_load_b128 -> LDS double buffers.
__global__ void __launch_bounds__(256, 2)
dfn_conv_wmma_kernel(const float* __restrict__ inpt,
                     const float* __restrict__ wdyn,
                     const float* __restrict__ biasbuf,
                     float* __restrict__ out) {
    __shared__ __align__(16) float wlds[NPp];                 // [36][16][2] weights
    __shared__ __align__(16) float win[8][2][3 * 18 * CINc];  // per-wave dbl-buf halo

    const int b    = blockIdx.y;
    const int wave = threadIdx.x >> 5;
    const int lane = threadIdx.x & 31;
    const int tg   = lane >> 4;   // lane half: K-slot pair select
    const int n    = lane & 15;   // A row (M) and B column (N)

    // ---- TDM: DMA the sample's 72x16 weight block (4608 B) into LDS ----
    if (wave == 0) {
        const unsigned long long ga =
            (unsigned long long)(const void*)(wdyn + (size_t)b * NPp);
        const unsigned int lds_off =
            (unsigned int)(unsigned long long)(void*)wlds;  // low 32b = LDS byte off
        // D# group 0: count=1 | lds_addr | global_addr[56:0] | type=2
        v4u g0 = {1u, lds_off, (unsigned int)ga,
                  (unsigned int)((ga >> 32) & 0x01FFFFFFu) | 0x80000000u};
        // D# group 1: data_size=4B, tensor_dim0=tile_dim0=1152 (1-D tile),
        // tensor_dim1=1, tile_dim1=0 (unused), dim0 stride = 1152.
        v8i g1 = {(int)(2u << 16),
                  (int)((NPp & 0xFFFFu) << 16),
                  (int)(1u << 16),
                  (int)((unsigned)NPp << 16),
                  0,
                  (int)NPp,
                  0, 0};
        v4i gz4 = {0, 0, 0, 0};
        v8i gz8 = {0, 0, 0, 0, 0, 0, 0, 0};
        __builtin_amdgcn_tensor_load_to_lds(g0, g1, gz4, gz4, gz8, 0);
    }

    const float biasn = biasbuf[b * NCc + n];
    const int tile0 = blockIdx.x * 64 + wave * 8;

    // ---- stage halo window for a tile into a buffer (float4 over channels) ----
    auto stage = [&](int tile, int buf) {
        const int pix0 = tile * 16;
        const int h0 = pix0 / Wn;
        const int c0 = pix0 - h0 * Wn;
        for (int i = lane; i < 3 * 18 * 2; i += 32) {
            const int cin4 = i & 1;
            const int col  = (i >> 1) % 18;
            const int kh   = i / 36;
            const int hh = h0 + kh - 1;
            const int ww = c0 + col - 1;
            float4 v = make_float4(0.f, 0.f, 0.f, 0.f);
            if (hh >= 0 && hh < Hh && ww >= 0 && ww < Wn) {
                v = *(const float4*)(inpt +
                    (((size_t)b * Hh + hh) * Wn + ww) * CINc + cin4 * 4);
            }
            ((float4*)win[wave][buf])[i] = v;
        }
    };

    stage(tile0, 0);                               // overlaps with the TDM DMA
    if (wave == 0) __builtin_amdgcn_s_wait_tensorcnt(0);
    __syncthreads();   // weights + window buf0 visible to all waves

    // ---- preload B fragments (constant across all tiles of this sample) ----
    v2f bfr[18];
    {
        const float* bp = wlds + (tg * NCc + n) * 2;   // [kpair][c][klo]
#pragma unroll
        for (int kk = 0; kk < 18; ++kk)
            bfr[kk] = *(const v2f*)(bp + kk * 64);     // ds_load_b64, imm offset
    }

    for (int t = 0; t < 8; ++t) {
        const int buf = t & 1;
        const int pix0 = (tile0 + t) * 16;

        // ---- 16x16x72 GEMM: 18 x (ds_load_b64 + v_wmma_f32_16x16x4_f32) ----
        const float* aptr = win[wave][buf] + n * CINc + tg * 2;
        v8f acc = {0.f, 0.f, 0.f, 0.f, 0.f, 0.f, 0.f, 0.f};
#pragma unroll
        for (int kk = 0; kk < 18; ++kk) {
            const v2f a = *(const v2f*)(aptr + aoff(kk));
            acc = __builtin_amdgcn_wmma_f32_16x16x4_f32(
                false, a, false, bfr[kk], (short)0, acc, false, false);
        }

        // stage next tile's window into the other buffer while WMMAs drain
        if (t < 7) stage(tile0 + t + 1, buf ^ 1);

        // epilogue: per-sample bias + ELU + coalesced store
        float* op = out + ((size_t)b * HW + pix0) * NCc + n;
#pragma unroll
        for (int v = 0; v < 8; ++v) {
            const int m = v + tg * 8;   // C/D layout: M = vgpr + 8*(lane>>4)
            op[m * NCc] = elu1(acc[v] + biasn);
        }
        __syncthreads();
    }
}

extern "C" void kernel_launch(void* const* d_in, const int* in_sizes, int n_in,
                              void* d_out, int out_size, void* d_ws, size_t ws_size,
                              hipStream_t stream) {
    (void)in_sizes; (void)n_in; (void)out_size; (void)ws_size;
    const float* inpt = (const float*)d_in[0];
    const float* fin  = (const float*)d_in[1];
    const float* W1   = (const float*)d_in[2];
    const float* b1   = (const float*)d_in[3];
    const float* Wd   = (const float*)d_in[4];
    const float* bd   = (const float*)d_in[5];
    const float* Wb   = (const float*)d_in[6];
    const float* bb   = (const float*)d_in[7];
    const float* dfnb = (const float*)d_in[8];
    float* out = (float*)d_out;

    float* ws      = (float*)d_ws;
    float* h       = ws;                     // 4096 floats
    float* wdyn    = ws + 4096;              // 18432 floats
    float* biasbuf = ws + 4096 + Bn * NPp;   // 256 floats

    dfn_param_kernel<<<16, 256, 0, stream>>>(fin, W1, b1, h);
    dfn_dyn_kernel<<<73, 256, 0, stream>>>(h, Wd, bd, Wb, bb, dfnb, wdyn, biasbuf);
    dfn_conv_wmma_kernel<<<dim3(49, 16), 256, 0, stream>>>(inpt, wdyn, biasbuf, out);
}